// My_UpSampling_IDWT_3D_41626823033439
// MI455X (gfx1250) — compile-verified
//
#include <hip/hip_runtime.h>
#include <cstdint>

// ---------------------------------------------------------------------------
// 3D inverse Haar DWT, butterfly form.
// out[n, c, 2d+pd, 2h+ph, 2w+pw] = s^3 * sum_{abc} sign * S_abc[n,c,d,h,w]
// subband name letters map: 1st->D axis, 2nd->H axis, 3rd->W axis.
// s^3 = 2^(-1.5)
// ---------------------------------------------------------------------------
#define S3 0.35355339059327378f

__global__ __launch_bounds__(256) void idwt3d_kernel(
    const float* __restrict__ pLLL, const float* __restrict__ pLLH,
    const float* __restrict__ pLHL, const float* __restrict__ pLHH,
    const float* __restrict__ pHLL, const float* __restrict__ pHLH,
    const float* __restrict__ pHHL, const float* __restrict__ pHHH,
    float* __restrict__ out)
{
    // Total threads = 2*64*32*32*8 = 1,048,576. Each handles 4 input w's.
    const unsigned tid = blockIdx.x * 256u + threadIdx.x;
    const unsigned wq = tid & 7u;          // w-quad: inputs 4wq..4wq+3
    const unsigned h  = (tid >> 3) & 31u;
    const unsigned d  = (tid >> 8) & 31u;
    const unsigned nc = tid >> 13;         // n*64 + c, in [0,128)

    const size_t inIdx = (size_t)tid * 4u; // same linear layout in every subband

    float4 a[8];
    a[0] = *(const float4*)(pLLL + inIdx);
    a[1] = *(const float4*)(pLLH + inIdx);
    a[2] = *(const float4*)(pLHL + inIdx);
    a[3] = *(const float4*)(pLHH + inIdx);
    a[4] = *(const float4*)(pHLL + inIdx);
    a[5] = *(const float4*)(pHLH + inIdx);
    a[6] = *(const float4*)(pHHL + inIdx);
    a[7] = *(const float4*)(pHHH + inIdx);

    float o[2][2][8];   // [pd][ph][2*j + pw], j = input w element 0..3
#pragma unroll
    for (int j = 0; j < 4; ++j) {
        const float lll = reinterpret_cast<const float*>(&a[0])[j];
        const float llh = reinterpret_cast<const float*>(&a[1])[j];
        const float lhl = reinterpret_cast<const float*>(&a[2])[j];
        const float lhh = reinterpret_cast<const float*>(&a[3])[j];
        const float hll = reinterpret_cast<const float*>(&a[4])[j];
        const float hlh = reinterpret_cast<const float*>(&a[5])[j];
        const float hhl = reinterpret_cast<const float*>(&a[6])[j];
        const float hhh = reinterpret_cast<const float*>(&a[7])[j];

        // D-axis combine (1st letter): pd=0 -> +, pd=1 -> -
        const float t0ll = lll + hll, t1ll = lll - hll;
        const float t0lh = llh + hlh, t1lh = llh - hlh;
        const float t0hl = lhl + hhl, t1hl = lhl - hhl;
        const float t0hh = lhh + hhh, t1hh = lhh - hhh;
        // H-axis combine (2nd letter)
        const float u00l = t0ll + t0hl, u01l = t0ll - t0hl;
        const float u00h = t0lh + t0hh, u01h = t0lh - t0hh;
        const float u10l = t1ll + t1hl, u11l = t1ll - t1hl;
        const float u10h = t1lh + t1hh, u11h = t1lh - t1hh;
        // W-axis combine (3rd letter) + scale
        o[0][0][2*j]   = S3 * (u00l + u00h);
        o[0][0][2*j+1] = S3 * (u00l - u00h);
        o[0][1][2*j]   = S3 * (u01l + u01h);
        o[0][1][2*j+1] = S3 * (u01l - u01h);
        o[1][0][2*j]   = S3 * (u10l + u10h);
        o[1][0][2*j+1] = S3 * (u10l - u10h);
        o[1][1][2*j]   = S3 * (u11l + u11h);
        o[1][1][2*j+1] = S3 * (u11l - u11h);
    }

    // Output channel: rec occupies channels [0,64) of each n in a 128-channel tensor.
    const unsigned ncOut = nc + (nc & 64u);  // n*128 + c
    const size_t base = (size_t)ncOut * 262144u
                      + (size_t)(2u * d) * 4096u
                      + (size_t)(2u * h) * 64u
                      + (size_t)(8u * wq);
#pragma unroll
    for (int pd = 0; pd < 2; ++pd) {
#pragma unroll
        for (int ph = 0; ph < 2; ++ph) {
            float* op = out + base + (size_t)pd * 4096u + (size_t)ph * 64u;
            *(float4*)(op)     = make_float4(o[pd][ph][0], o[pd][ph][1],
                                             o[pd][ph][2], o[pd][ph][3]);
            *(float4*)(op + 4) = make_float4(o[pd][ph][4], o[pd][ph][5],
                                             o[pd][ph][6], o[pd][ph][7]);
        }
    }
}

// ---------------------------------------------------------------------------
// feature_map -> out channels [64,128) copy, staged via the CDNA5 async
// global->LDS data mover. Only async *loads* use ASYNCcnt, and async loads
// complete in order, so an 8-slot pipeline with a constant
// `s_wait_asynccnt 7` keeps 8 loads in flight per wave. Stores are normal
// pipelined global_store_b128 (STOREcnt).
// 1024 blocks x 256 threads; 8,388,608 float4 total = 32 per thread, exact.
// ---------------------------------------------------------------------------
#define CP_T (1024u * 256u)   // total threads

__global__ __launch_bounds__(256) void concat_copy_async(
    const float4* __restrict__ src, float4* __restrict__ dst)
{
    __shared__ float4 buf[8][256];   // 32 KB: 8 slots x 256 lanes x 16B
    const unsigned tid  = blockIdx.x * 256u + threadIdx.x;
    // Low 32 bits of a generic shared pointer == allocation-relative LDS offset.
    const unsigned lds0 = (unsigned)(uintptr_t)&buf[0][threadIdx.x];

    // Prologue: fill all 8 slots (batch 0).
#pragma unroll
    for (int s = 0; s < 8; ++s) {
        unsigned long long ga =
            (unsigned long long)(uintptr_t)(src + ((size_t)s * CP_T + tid));
        unsigned la = lds0 + (unsigned)(s * 4096);
        asm volatile("global_load_async_to_lds_b128 %0, %1, off"
                     :: "v"(la), "v"(ga) : "memory");
    }

    // Steady state: batches 0..2 consumed, batches 1..3 refilled.
#pragma unroll
    for (int b = 0; b < 3; ++b) {
#pragma unroll
        for (int s = 0; s < 8; ++s) {
            // issued = 8 + b*8 + s; need load #(b*8+s+1) done -> outstanding <= 7
            asm volatile("s_wait_asynccnt 7" ::: "memory");
            float4 v = buf[s][threadIdx.x];
            size_t q  = (size_t)(b * 8 + s) * CP_T + tid;           // src float4 idx
            dst[q + ((q >> 22) + 1) * 4194304ull] = v;              // channel shift
            // Refill this slot with the next batch's element.
            unsigned long long ga = (unsigned long long)(uintptr_t)
                (src + ((size_t)((b + 1) * 8 + s) * CP_T + tid));
            unsigned la = lds0 + (unsigned)(s * 4096);
            asm volatile("global_load_async_to_lds_b128 %0, %1, off"
                         :: "v"(la), "v"(ga) : "memory");
        }
    }

    // Epilogue: drain batch 3.
    asm volatile("s_wait_asynccnt 0" ::: "memory");
#pragma unroll
    for (int s = 0; s < 8; ++s) {
        float4 v = buf[s][threadIdx.x];
        size_t q = (size_t)(24 + s) * CP_T + tid;
        dst[q + ((q >> 22) + 1) * 4194304ull] = v;
    }
}

extern "C" void kernel_launch(void* const* d_in, const int* in_sizes, int n_in,
                              void* d_out, int out_size, void* d_ws, size_t ws_size,
                              hipStream_t stream) {
    (void)in_sizes; (void)n_in; (void)out_size; (void)d_ws; (void)ws_size;
    const float* pLLL = (const float*)d_in[0];
    const float* pLLH = (const float*)d_in[1];
    const float* pLHL = (const float*)d_in[2];
    const float* pLHH = (const float*)d_in[3];
    const float* pHLL = (const float*)d_in[4];
    const float* pHLH = (const float*)d_in[5];
    const float* pHHL = (const float*)d_in[6];
    const float* pHHH = (const float*)d_in[7];
    const float* feat = (const float*)d_in[8];
    float* out = (float*)d_out;

    // 2*64*32*32*8 threads / 256 = 4096 blocks
    idwt3d_kernel<<<4096, 256, 0, stream>>>(pLLL, pLLH, pLHL, pLHH,
                                            pHLL, pHLH, pHHL, pHHH, out);
    // feature map: 33,554,432 floats = 8,388,608 float4
    concat_copy_async<<<1024, 256, 0, stream>>>((const float4*)feat,
                                                (float4*)out);
}